// SSA_50921132262136
// MI455X (gfx1250) — compile-verified
//
#include <hip/hip_runtime.h>
#include <hip/hip_bf16.h>
#include <math.h>

// ---------------- problem constants ----------------
#define Bb   4
#define Cc   64
#define Tt   8
#define Hh   32
#define Ww   32
#define Oo   64
#define HW   1024          // H*W
#define Nn   32            // B*T
#define CHW  65536         // C*HW

// workspace layout (float offsets)
#define OFF_Q   ((size_t)0)
#define OFF_K   ((size_t)2097152)
#define OFF_V   ((size_t)4194304)
#define OFF_A   ((size_t)6291456)   // spatial-attn output (n,c,hw)
#define OFF_XE  ((size_t)8388608)   // tconv output (b,t,o,hw)
#define OFF_SC  ((size_t)10485760)  // channel logits (32,64,64)
#define OFF_MT  ((size_t)10616832)  // temporal logits (4,8,8)
#define OFF_SS  ((size_t)10617088)  // spatial logits (32,1024,1024) ~134MB (L2-resident)

typedef __attribute__((ext_vector_type(2))) float v2f;
typedef __attribute__((ext_vector_type(8))) float v8f;

__device__ __forceinline__ v8f wmma4(v2f a, v2f b, v8f c) {
  // D(16x16,f32) = A(16x4,f32) * B(4x16,f32) + C
  return __builtin_amdgcn_wmma_f32_16x16x4_f32(false, a, false, b, (short)0, c,
                                               false, false);
}

__device__ __forceinline__ v8f zero8() {
  v8f z = {0.f,0.f,0.f,0.f,0.f,0.f,0.f,0.f};
  return z;
}

// =====================================================================
// K1: QKV projection.  y[o,i] = relu(sum_c W[o,c]*x[c,i] + b[o])
// per (b,t) slice; x element (c,i) at xbase + c*(T*HW) + i.
// grid (4 colblocks, 32 slices), block 256 (8 waves).
// =====================================================================
__global__ __launch_bounds__(256) void k_qkv(
    const float* __restrict__ x,
    const float* __restrict__ wq, const float* __restrict__ bq,
    const float* __restrict__ wk, const float* __restrict__ bk,
    const float* __restrict__ wv, const float* __restrict__ bv,
    float* __restrict__ q, float* __restrict__ k, float* __restrict__ v) {
  __shared__ float w_s[3 * 64 * 64];
  __shared__ float b_s[3 * 64];
  const int tid = threadIdx.x;
  for (int idx = tid; idx < 4096; idx += 256) {
    w_s[idx]        = wq[idx];
    w_s[4096 + idx] = wk[idx];
    w_s[8192 + idx] = wv[idx];
  }
  if (tid < 64) { b_s[tid] = bq[tid]; b_s[64 + tid] = bk[tid]; b_s[128 + tid] = bv[tid]; }
  __syncthreads();

  const int n = blockIdx.y;
  const int bb = n >> 3, tt = n & 7;
  const float* xbase = x + ((size_t)bb * Cc * Tt + tt) * HW;  // +c*T*HW+i
  const int j0 = blockIdx.x * 256;
  const int wave = tid >> 5, lane = tid & 31;
  const int half = lane >> 4, l16 = lane & 15;

  float* outp[3] = {q + (size_t)n * CHW, k + (size_t)n * CHW, v + (size_t)n * CHW};

  for (int ct = 0; ct < 2; ++ct) {
    const int n0 = j0 + (wave * 2 + ct) * 16;
    v8f acc[3][4];
    #pragma unroll
    for (int m = 0; m < 3; ++m)
      #pragma unroll
      for (int rt = 0; rt < 4; ++rt) acc[m][rt] = zero8();

    #pragma unroll 4
    for (int k0 = 0; k0 < 64; k0 += 4) {
      v2f bf;
      bf.x = xbase[(size_t)(k0 + 2 * half) * (Tt * HW) + n0 + l16];
      bf.y = xbase[(size_t)(k0 + 2 * half + 1) * (Tt * HW) + n0 + l16];
      #pragma unroll
      for (int rt = 0; rt < 4; ++rt) {
        #pragma unroll
        for (int m = 0; m < 3; ++m) {
          const int abase = m * 4096 + (rt * 16 + l16) * 64 + k0 + 2 * half;
          v2f af; af.x = w_s[abase]; af.y = w_s[abase + 1];
          acc[m][rt] = wmma4(af, bf, acc[m][rt]);
        }
      }
    }
    #pragma unroll
    for (int m = 0; m < 3; ++m)
      #pragma unroll
      for (int rt = 0; rt < 4; ++rt)
        #pragma unroll
        for (int r = 0; r < 8; ++r) {
          const int row = rt * 16 + r + 8 * half;
          float val = acc[m][rt][r] + b_s[m * 64 + row];
          val = fmaxf(val, 0.f);
          outp[m][(size_t)row * HW + n0 + l16] = val;
        }
  }
}

// =====================================================================
// K2: channel logits  Sc[n][c,d] = sum_i Q[c,i]*K[d,i]   (K=1024)
// grid 32, block 128 (4 waves, wave = c row-tile)
// =====================================================================
__global__ __launch_bounds__(128) void k_sc(const float* __restrict__ q,
                                            const float* __restrict__ kk,
                                            float* __restrict__ sc) {
  const int n = blockIdx.x;
  const float* Q = q + (size_t)n * CHW;
  const float* K = kk + (size_t)n * CHW;
  const int tid = threadIdx.x, wave = tid >> 5, lane = tid & 31;
  const int half = lane >> 4, l16 = lane & 15;
  const int m0 = wave * 16;
  v8f acc[4];
  #pragma unroll
  for (int i = 0; i < 4; ++i) acc[i] = zero8();

  for (int k0 = 0; k0 < HW; k0 += 4) {
    v2f af;
    af.x = Q[(size_t)(m0 + l16) * HW + k0 + 2 * half];
    af.y = Q[(size_t)(m0 + l16) * HW + k0 + 2 * half + 1];
    #pragma unroll
    for (int dt = 0; dt < 4; ++dt) {
      v2f bf;
      bf.x = K[(size_t)(dt * 16 + l16) * HW + k0 + 2 * half];
      bf.y = K[(size_t)(dt * 16 + l16) * HW + k0 + 2 * half + 1];
      acc[dt] = wmma4(af, bf, acc[dt]);
    }
  }
  #pragma unroll
  for (int dt = 0; dt < 4; ++dt)
    #pragma unroll
    for (int r = 0; r < 8; ++r) {
      const int row = m0 + r + 8 * half;
      sc[(size_t)n * 4096 + row * 64 + dt * 16 + l16] = acc[dt][r];
    }
}

// =====================================================================
// K3: spatial logits Ss[n][i,j] = sum_c Q[c,i]*K[c,j]   (K=64)
// grid (64 blocks of 128x128, 32 n), block 256 (8 waves; wave = i row-tile)
// =====================================================================
__global__ __launch_bounds__(256) void k_ss(const float* __restrict__ q,
                                            const float* __restrict__ kk,
                                            float* __restrict__ ss) {
  const int n = blockIdx.y;
  const int i0 = (blockIdx.x >> 3) * 128;
  const int j0 = (blockIdx.x & 7) * 128;
  const float* Q = q + (size_t)n * CHW;
  const float* K = kk + (size_t)n * CHW;
  const int tid = threadIdx.x, wave = tid >> 5, lane = tid & 31;
  const int half = lane >> 4, l16 = lane & 15;
  const int m0 = i0 + wave * 16;
  v8f acc[8];
  #pragma unroll
  for (int i = 0; i < 8; ++i) acc[i] = zero8();

  #pragma unroll 4
  for (int k0 = 0; k0 < 64; k0 += 4) {
    v2f af;   // A[i][c] = Q[c][i]  (transposed read, lanes contiguous in i)
    af.x = Q[(size_t)(k0 + 2 * half) * HW + m0 + l16];
    af.y = Q[(size_t)(k0 + 2 * half + 1) * HW + m0 + l16];
    #pragma unroll
    for (int jt = 0; jt < 8; ++jt) {
      const int col = j0 + jt * 16 + l16;
      v2f bf;
      bf.x = K[(size_t)(k0 + 2 * half) * HW + col];
      bf.y = K[(size_t)(k0 + 2 * half + 1) * HW + col];
      acc[jt] = wmma4(af, bf, acc[jt]);
    }
  }
  float* out = ss + (size_t)n * HW * HW;
  #pragma unroll
  for (int jt = 0; jt < 8; ++jt)
    #pragma unroll
    for (int r = 0; r < 8; ++r) {
      const int row = m0 + r + 8 * half;
      out[(size_t)row * HW + j0 + jt * 16 + l16] = acc[jt][r];
    }
}

// =====================================================================
// K4/K5: softmax over the n axis (axis=0), in place.
// =====================================================================
__global__ __launch_bounds__(256) void k_softmax_n(float* __restrict__ buf,
                                                   int stride, int nslices) {
  const int pos = blockIdx.x * 256 + threadIdx.x;
  if (pos >= stride) return;
  float vals[32];
  float mx = -3.4e38f;
  for (int nn = 0; nn < nslices; ++nn) {
    float v = buf[(size_t)nn * stride + pos];
    vals[nn] = v;
    mx = fmaxf(mx, v);
  }
  float s = 0.f;
  for (int nn = 0; nn < nslices; ++nn) {
    float e = __expf(vals[nn] - mx);
    vals[nn] = e;
    s += e;
  }
  const float inv = 1.f / s;
  for (int nn = 0; nn < nslices; ++nn)
    buf[(size_t)nn * stride + pos] = vals[nn] * inv;
}

// =====================================================================
// K6a: a1[n][c,i] = sum_d mc[c,d]*V[d,i]   (K=64) -> writes A
// grid (4 colblocks, 32 n), block 128 (4 waves; wave owns 4 col-tiles)
// =====================================================================
__global__ __launch_bounds__(128) void k_ac(const float* __restrict__ sc,
                                            const float* __restrict__ v,
                                            float* __restrict__ a) {
  const int n = blockIdx.y;
  const float* MC = sc + (size_t)n * 4096;
  const float* V = v + (size_t)n * CHW;
  const int j0 = blockIdx.x * 256;
  const int tid = threadIdx.x, wave = tid >> 5, lane = tid & 31;
  const int half = lane >> 4, l16 = lane & 15;
  v8f acc[4][4];
  #pragma unroll
  for (int rt = 0; rt < 4; ++rt)
    #pragma unroll
    for (int jt = 0; jt < 4; ++jt) acc[rt][jt] = zero8();

  #pragma unroll 4
  for (int k0 = 0; k0 < 64; k0 += 4) {
    v2f bf[4];
    #pragma unroll
    for (int jt = 0; jt < 4; ++jt) {
      const int col = j0 + (wave * 4 + jt) * 16 + l16;
      bf[jt].x = V[(size_t)(k0 + 2 * half) * HW + col];
      bf[jt].y = V[(size_t)(k0 + 2 * half + 1) * HW + col];
    }
    #pragma unroll
    for (int rt = 0; rt < 4; ++rt) {
      v2f af;
      af.x = MC[(rt * 16 + l16) * 64 + k0 + 2 * half];
      af.y = MC[(rt * 16 + l16) * 64 + k0 + 2 * half + 1];
      #pragma unroll
      for (int jt = 0; jt < 4; ++jt) acc[rt][jt] = wmma4(af, bf[jt], acc[rt][jt]);
    }
  }
  float* out = a + (size_t)n * CHW;
  #pragma unroll
  for (int rt = 0; rt < 4; ++rt)
    #pragma unroll
    for (int jt = 0; jt < 4; ++jt)
      #pragma unroll
      for (int r = 0; r < 8; ++r) {
        const int row = rt * 16 + r + 8 * half;
        const int col = j0 + (wave * 4 + jt) * 16 + l16;
        out[(size_t)row * HW + col] = acc[rt][jt][r];
      }
}

// =====================================================================
// K6b: a[n][c,i] += sum_j V[c,j]*ms[i,j]   (K=1024)
// LDS-staged: both the 128-i-wide ms block and the V rows are loaded
// coalesced from global in 64-wide j chunks (ms transposed on store) into
// padded LDS (129/65-float rows -> conflict-free stores AND frag reads).
// Each WG reads its ms block exactly once (one total pass over 134MB).
// grid (8 iblocks of 128, 32 n), block 128 (4 waves; wave = c row-tile)
// =====================================================================
__global__ __launch_bounds__(128) void k_as(const float* __restrict__ ss,
                                            const float* __restrict__ v,
                                            float* __restrict__ a) {
  __shared__ float ms_s[64 * 129];  // [j_local][i_local], pad 129 -> bank (j+i)%64
  __shared__ float v_s[64 * 65];    // [c_local][j_local],  pad 65  -> bank (c+j)%64
  const int n = blockIdx.y;
  const float* MS = ss + (size_t)n * HW * HW;
  const float* V = v + (size_t)n * CHW;
  const int i0 = blockIdx.x * 128;
  const int tid = threadIdx.x, wave = tid >> 5, lane = tid & 31;
  const int half = lane >> 4, l16 = lane & 15;
  const int m0 = wave * 16;
  const int ldj = tid & 63;        // j_local for cooperative loads
  const int ldh = tid >> 6;        // 0/1 row split
  v8f acc[8];
  #pragma unroll
  for (int i = 0; i < 8; ++i) acc[i] = zero8();

  for (int kc = 0; kc < 16; ++kc) {
    __syncthreads();
    // stage ms block: 128 i-rows x 64 j, coalesced reads, transposed store
    #pragma unroll 8
    for (int it = 0; it < 64; ++it) {
      const int il = it * 2 + ldh;
      ms_s[ldj * 129 + il] = MS[(size_t)(i0 + il) * HW + kc * 64 + ldj];
    }
    // stage V chunk: 64 c-rows x 64 j
    #pragma unroll 8
    for (int it = 0; it < 32; ++it) {
      const int cl = it * 2 + ldh;
      v_s[cl * 65 + ldj] = V[(size_t)cl * HW + kc * 64 + ldj];
    }
    if (kc + 1 < 16)   // gfx1250 global_prefetch_b8 for the next chunk
      __builtin_prefetch(&MS[(size_t)(i0 + tid) * HW + (kc + 1) * 64], 0, 1);
    __syncthreads();

    #pragma unroll 4
    for (int kl = 0; kl < 64; kl += 4) {
      v2f af;   // A[c][j] = V[c][j]
      af.x = v_s[(m0 + l16) * 65 + kl + 2 * half];
      af.y = v_s[(m0 + l16) * 65 + kl + 2 * half + 1];
      #pragma unroll
      for (int jt = 0; jt < 8; ++jt) {
        v2f bf;   // B[j][i] = ms[i][j] (already transposed in LDS)
        bf.x = ms_s[(kl + 2 * half) * 129 + jt * 16 + l16];
        bf.y = ms_s[(kl + 2 * half + 1) * 129 + jt * 16 + l16];
        acc[jt] = wmma4(af, bf, acc[jt]);
      }
    }
  }
  float* out = a + (size_t)n * CHW;
  #pragma unroll
  for (int jt = 0; jt < 8; ++jt)
    #pragma unroll
    for (int r = 0; r < 8; ++r) {
      const int row = m0 + r + 8 * half;
      const size_t idx = (size_t)row * HW + i0 + jt * 16 + l16;
      out[idx] += acc[jt][r];
    }
}

// =====================================================================
// K7: temporal conv(3,1,1) + bias + relu.
// Xe[(b,t)][o,i] = relu(sum_{dk} sum_c wX[o,c,dk]*A[(b,t+dk-1)][c,i] + bX[o])
// grid (4 colblocks, 32 n), block 256 (8 waves)
// =====================================================================
__global__ __launch_bounds__(256) void k_tconv(const float* __restrict__ a,
                                               const float* __restrict__ wX,
                                               const float* __restrict__ bX,
                                               float* __restrict__ xe) {
  __shared__ float w_s[3 * 64 * 64];   // [dk][o][c]
  __shared__ float b_s[64];
  const int tid = threadIdx.x;
  for (int idx = tid; idx < 4096; idx += 256) {
    #pragma unroll
    for (int dk = 0; dk < 3; ++dk) w_s[dk * 4096 + idx] = wX[idx * 3 + dk];
  }
  if (tid < 64) b_s[tid] = bX[tid];
  __syncthreads();

  const int n = blockIdx.y;
  const int bb = n >> 3, tt = n & 7;
  const int j0 = blockIdx.x * 256;
  const int wave = tid >> 5, lane = tid & 31;
  const int half = lane >> 4, l16 = lane & 15;

  for (int ct = 0; ct < 2; ++ct) {
    const int n0 = j0 + (wave * 2 + ct) * 16;
    v8f acc[4];
    #pragma unroll
    for (int rt = 0; rt < 4; ++rt) acc[rt] = zero8();

    for (int dk = 0; dk < 3; ++dk) {     // uniform across block -> EXEC stays full
      const int ts = tt + dk - 1;
      if (ts < 0 || ts >= Tt) continue;
      const float* S = a + ((size_t)bb * Tt + ts) * CHW;
      #pragma unroll 4
      for (int k0 = 0; k0 < 64; k0 += 4) {
        v2f bf;
        bf.x = S[(size_t)(k0 + 2 * half) * HW + n0 + l16];
        bf.y = S[(size_t)(k0 + 2 * half + 1) * HW + n0 + l16];
        #pragma unroll
        for (int rt = 0; rt < 4; ++rt) {
          const int abase = dk * 4096 + (rt * 16 + l16) * 64 + k0 + 2 * half;
          v2f af; af.x = w_s[abase]; af.y = w_s[abase + 1];
          acc[rt] = wmma4(af, bf, acc[rt]);
        }
      }
    }
    float* out = xe + (size_t)n * CHW;
    #pragma unroll
    for (int rt = 0; rt < 4; ++rt)
      #pragma unroll
      for (int r = 0; r < 8; ++r) {
        const int row = rt * 16 + r + 8 * half;
        float val = acc[rt][r] + b_s[row];
        out[(size_t)row * HW + n0 + l16] = fmaxf(val, 0.f);
      }
  }
}

// =====================================================================
// K8: temporal logits L[b,t,s] = sum_d Xe[b,t,d]*Xe[b,s,d], d=65536
// grid (64 pairs, 4 b), block 256, LDS tree reduce.
// =====================================================================
__global__ __launch_bounds__(256) void k_tq(const float* __restrict__ xe,
                                            float* __restrict__ L) {
  __shared__ float red[256];
  const int bb = blockIdx.y;
  const int tt = blockIdx.x >> 3, ss = blockIdx.x & 7;
  const float* Xq = xe + ((size_t)bb * Tt + tt) * CHW;
  const float* Xs = xe + ((size_t)bb * Tt + ss) * CHW;
  float p = 0.f;
  for (int d = threadIdx.x; d < CHW; d += 256) p += Xq[d] * Xs[d];
  red[threadIdx.x] = p;
  __syncthreads();
  for (int s = 128; s > 0; s >>= 1) {
    if (threadIdx.x < s) red[threadIdx.x] += red[threadIdx.x + s];
    __syncthreads();
  }
  if (threadIdx.x == 0) L[bb * 64 + blockIdx.x] = red[0];
}

// =====================================================================
// K10: out[b,c,t,i] = sum_s mt[b,t,s] * V[(b,s)][c,i]
// idx = b*2^19 + c*2^13 + t*2^10 + i  == output flat index (b,c,t,h,w)
// =====================================================================
__global__ __launch_bounds__(256) void k_out(const float* __restrict__ mt,
                                             const float* __restrict__ v,
                                             float* __restrict__ out) {
  const int idx = blockIdx.x * 256 + threadIdx.x;
  const int i = idx & 1023;
  const int t = (idx >> 10) & 7;
  const int c = (idx >> 13) & 63;
  const int b = idx >> 19;
  float s = 0.f;
  #pragma unroll
  for (int ss = 0; ss < 8; ++ss) {
    const float w = mt[b * 64 + t * 8 + ss];
    s += w * v[(((size_t)b * 8 + ss) * 64 + c) * HW + i];
  }
  out[idx] = s;
}

// =====================================================================
extern "C" void kernel_launch(void* const* d_in, const int* in_sizes, int n_in,
                              void* d_out, int out_size, void* d_ws, size_t ws_size,
                              hipStream_t stream) {
  (void)in_sizes; (void)n_in; (void)out_size; (void)ws_size;
  const float* x  = (const float*)d_in[0];
  const float* wq = (const float*)d_in[1];
  const float* bq = (const float*)d_in[2];
  const float* wk = (const float*)d_in[3];
  const float* bk = (const float*)d_in[4];
  const float* wv = (const float*)d_in[5];
  const float* bv = (const float*)d_in[6];
  const float* wX = (const float*)d_in[7];
  const float* bX = (const float*)d_in[8];
  float* out = (float*)d_out;

  float* ws   = (float*)d_ws;
  float* wsQ  = ws + OFF_Q;
  float* wsK  = ws + OFF_K;
  float* wsV  = ws + OFF_V;
  float* wsA  = ws + OFF_A;
  float* wsXE = ws + OFF_XE;
  float* wsSC = ws + OFF_SC;
  float* wsMT = ws + OFF_MT;
  float* wsSS = ws + OFF_SS;

  // 1) QKV projections (+bias, relu)
  k_qkv<<<dim3(4, Nn), 256, 0, stream>>>(x, wq, bq, wk, bk, wv, bv, wsQ, wsK, wsV);
  // 2) channel logits
  k_sc<<<dim3(Nn), 128, 0, stream>>>(wsQ, wsK, wsSC);
  // 3) spatial logits (L2-resident 134MB)
  k_ss<<<dim3(64, Nn), 256, 0, stream>>>(wsQ, wsK, wsSS);
  // 4) softmax over n for channel logits (stride 64*64)
  k_softmax_n<<<dim3(16), 256, 0, stream>>>(wsSC, 4096, Nn);
  // 5) softmax over n for spatial logits (stride 1024*1024)
  k_softmax_n<<<dim3(4096), 256, 0, stream>>>(wsSS, HW * HW, Nn);
  // 6) a = m_c@V  then  a += V@m_s^T
  k_ac<<<dim3(4, Nn), 128, 0, stream>>>(wsSC, wsV, wsA);
  k_as<<<dim3(8, Nn), 128, 0, stream>>>(wsSS, wsV, wsA);
  // 7) temporal conv + bias + relu
  k_tconv<<<dim3(4, Nn), 256, 0, stream>>>(wsA, wX, bX, wsXE);
  // 8) temporal logits
  k_tq<<<dim3(64, Bb), 256, 0, stream>>>(wsXE, wsMT);
  // 9) softmax over b for temporal logits (stride 8*8)
  k_softmax_n<<<dim3(1), 256, 0, stream>>>(wsMT, 64, Bb);
  // 10) final weighted gather, writes (b,c,t,h,w) directly
  k_out<<<dim3(8192), 256, 0, stream>>>(wsMT, wsV, out);
}